// FeatureRelationDecoder_72834055406376
// MI455X (gfx1250) — compile-verified
//
#include <hip/hip_runtime.h>
#include <stdint.h>

// FeatureRelationDecoder: out[i,j,0:8] = pair_mask(i,j) ? z[i,:]*z[j,:] : [1,0,0,0,0,0,0,0]
// pair_mask = (seg[i,j] + (i==j)) == 0  &&  node_mask[i] && node_mask[j]  &&  batch[i]==batch[j]
//
// Memory-bound: 302 MB NT stores + 38 MB NT loads -> ~14.6 us floor at 23.3 TB/s.
// gfx1250 paths: global_load_async_to_lds_b128 (+ s_wait_asynccnt) stages the z
// j-tile in LDS once per block; the tile is then reused across 8 output rows
// (8x less L2->LDS traffic than one-row blocks). ds_load_b128 readback,
// non-temporal global_store_b128 for the never-re-read 302 MB output.

typedef float v4f __attribute__((ext_vector_type(4)));

#define THREADS 256
#define TILE_J  1024               // j columns per block (32 KB z tile in LDS)
#define ROWS_I  8                  // output rows per block sharing the tile
#define CHUNKS  (TILE_J / THREADS)
#define R       8

__global__ __launch_bounds__(THREADS)
void FeatureRelationDecoder_pair_kernel(const float* __restrict__ z,
                                        const int*   __restrict__ seg,
                                        const int*   __restrict__ cls,
                                        const int*   __restrict__ batch,
                                        float*       __restrict__ out,
                                        int n) {
    __shared__ __align__(16) float zj_tile[TILE_J * R];   // 32 KB per block

    const int t  = threadIdx.x;
    const int i0 = blockIdx.y * ROWS_I;  // first row of this block
    const int j0 = blockIdx.x * TILE_J;  // start of j tile

    // ---- Stage z rows of this j-tile into LDS via gfx1250 async copy ----
    // ISA 08 §4.4: the instruction offset is added to both the LDS and the memory address,
    // so one LDS-address VGPR + one 64-bit address pair serves both 16B halves.
#pragma unroll
    for (int c = 0; c < CHUNKS; ++c) {
        const int node = c * THREADS + t;
        const int j    = j0 + node;
        if (j < n) {
            // Low 32 bits of a generic LDS pointer are the wave-relative LDS
            // byte offset (aperture lives in the high word) -> valid VDST value.
            unsigned lds_off = (unsigned)(uintptr_t)(&zj_tile[node * R]);
            unsigned long long gaddr =
                (unsigned long long)(uintptr_t)(z + (size_t)j * R);
            asm volatile(
                "global_load_async_to_lds_b128 %0, %1, off\n\t"
                "global_load_async_to_lds_b128 %0, %1, off offset:16"
                :: "v"(lds_off), "v"(gaddr) : "memory");
        }
    }
    // Each lane only reads back what it staged itself -> per-wave wait suffices,
    // no workgroup barrier needed.
    asm volatile("s_wait_asynccnt 0x0" ::: "memory");

    const v4f one_hot = (v4f){1.0f, 0.0f, 0.0f, 0.0f};
    const v4f zeros   = (v4f){0.0f, 0.0f, 0.0f, 0.0f};

    // ---- 8 rows reuse the staged tile ----
#pragma unroll 1
    for (int i = i0; i < i0 + ROWS_I; ++i) {
        if (i >= n) break;

        // Block-uniform i-side values (i derived from blockIdx.y -> SGPRs)
        const int  ci     = cls[i];
        const int  bi     = batch[i];
        const bool mask_i = ((unsigned)(ci - 24) >= 3u);   // cls != 24,25,26
        const v4f  zi_lo  = *(const v4f*)(z + (size_t)i * R);
        const v4f  zi_hi  = *(const v4f*)(z + (size_t)i * R + 4);

#pragma unroll
        for (int c = 0; c < CHUNKS; ++c) {
            const int node = c * THREADS + t;
            const int j    = j0 + node;
            if (j >= n) continue;

            // seg_matrix streamed once -> non-temporal load (keeps L2 for z/cls/batch)
            int s = __builtin_nontemporal_load(&seg[(size_t)i * n + j]);
            s += (i == j);                                  // + eye(N)

            const int  cj     = cls[j];                     // 12 KB array: L1/L2 hot
            const int  bj     = batch[j];
            const bool mask_j = ((unsigned)(cj - 24) >= 3u);
            const bool p      = (s == 0) & mask_i & mask_j & (bi == bj);

            const v4f zj_lo = *(const v4f*)(&zj_tile[node * R]);   // ds_load_b128
            const v4f zj_hi = *(const v4f*)(&zj_tile[node * R + 4]);

            const v4f r0 = p ? (zi_lo * zj_lo) : one_hot;
            const v4f r1 = p ? (zi_hi * zj_hi) : zeros;

            // 302 MB output, never re-read -> non-temporal b128 stores
            float* o = out + ((size_t)i * n + j) * R;
            __builtin_nontemporal_store(r0, (v4f*)o);
            __builtin_nontemporal_store(r1, (v4f*)(o + 4));
        }
    }
}

extern "C" void kernel_launch(void* const* d_in, const int* in_sizes, int n_in,
                              void* d_out, int out_size, void* d_ws, size_t ws_size,
                              hipStream_t stream) {
    // setup_inputs() order: z [N,R] f32, seg_matrix [N,N] i32, cls_label [N] i32, batch [N] i32
    const float* z     = (const float*)d_in[0];
    const int*   seg   = (const int*)d_in[1];
    const int*   cls   = (const int*)d_in[2];
    const int*   batch = (const int*)d_in[3];
    float*       out   = (float*)d_out;

    const int n = in_sizes[2];   // NUM_NODES (= 3072)

    dim3 grid((n + TILE_J - 1) / TILE_J, (n + ROWS_I - 1) / ROWS_I);  // (3, 384)
    FeatureRelationDecoder_pair_kernel<<<grid, THREADS, 0, stream>>>(
        z, seg, cls, batch, out, n);
}